// MeGCN_35235911696847
// MI455X (gfx1250) — compile-verified
//
#include <hip/hip_runtime.h>
#include <hip/hip_bf16.h>
#include <math.h>

#define N_USERS 100000
#define N_ITEMS 40000
#define N_NODES (N_USERS + N_ITEMS)
#define DIM 64
#define ALPHA 0.2f

typedef float v2f __attribute__((ext_vector_type(2)));
typedef float v8f __attribute__((ext_vector_type(8)));

#define KC 128        // K-chunk staged through LDS
#define AP 132        // padded A row stride in LDS (floats); 132*4 % 16 == 0
#define WP 68         // padded W row stride in LDS (floats); 68*4  % 16 == 0

// ---------------------------------------------------------------------------
// C[16x64 slab] = l2norm_rows(A[16xK] * W[Kx64] + bias)
// 1 block = 128 threads = 4 waves; wave w computes 16x16 tile at cols 16w.
// Uses V_WMMA_F32_16X16X4_F32 (full fp32, matches reference precision).
// ---------------------------------------------------------------------------
__global__ __launch_bounds__(128) void gemm_bias_l2norm_wmma(
    const float* __restrict__ A,     // [M, K]
    const float* __restrict__ W,     // [K, 64]
    const float* __restrict__ bias,  // [64]
    float* __restrict__ out,         // [M, 64]
    int K)
{
    __shared__ float lds_a[16 * AP];     // 16 x KC tile of A (padded)
    __shared__ float lds_w[KC * WP];     // KC x 64 tile of W (padded)
    __shared__ float part[4][16];        // per-wave row sum-of-squares
    __shared__ float rscale[16];         // per-row 1/norm

    const int tid  = threadIdx.x;
    const int wave = tid >> 5;
    const int lane = tid & 31;
    const int half = lane >> 4;          // 0: lanes 0-15, 1: lanes 16-31
    const int nl   = lane & 15;
    const int n0   = wave * 16;          // this wave's column tile
    const int rowbase = blockIdx.x * 16;

    v8f acc = {};

    const int nchunks = K / KC;
    for (int ch = 0; ch < nchunks; ++ch) {
        const int kbase = ch * KC;
        __syncthreads();
        // A tile: 16 x 128 = 512 float4, 4 per thread, coalesced
        #pragma unroll
        for (int i = 0; i < 4; ++i) {
            int li = tid + i * 128;            // 0..511
            int r  = li >> 5;                  // row 0..15
            int c  = (li & 31) << 2;           // col 0..124 step 4
            float4 v = *(const float4*)(A + (size_t)(rowbase + r) * K + kbase + c);
            *(float4*)(&lds_a[r * AP + c]) = v;
        }
        // W tile: 128 x 64 = 2048 float4, 16 per thread, coalesced
        #pragma unroll
        for (int i = 0; i < 16; ++i) {
            int li = tid + i * 128;            // 0..2047
            int r  = li >> 4;                  // row 0..127
            int c  = (li & 15) << 2;           // col 0..60 step 4
            float4 v = *(const float4*)(W + (size_t)(kbase + r) * DIM + c);
            *(float4*)(&lds_w[r * WP + c]) = v;
        }
        __syncthreads();

        // ISA 7.12.2 f32 fragments:
        //  A 16x4:  VGPR0 = K{0|2}, VGPR1 = K{1|3} (halves of the wave)
        //  B 4x16:  VGPR0 = K-row{0|2}, VGPR1 = K-row{1|3}, N = lane&15
        #pragma unroll 8
        for (int kk = 0; kk < KC; kk += 4) {
            const int ka = kk + 2 * half;
            v2f a, b;
            a.x = lds_a[nl * AP + ka];
            a.y = lds_a[nl * AP + ka + 1];
            b.x = lds_w[ka * WP + n0 + nl];
            b.y = lds_w[(ka + 1) * WP + n0 + nl];
            acc = __builtin_amdgcn_wmma_f32_16x16x4_f32(
                /*neg_a=*/false, a, /*neg_b=*/false, b,
                /*c_mod=*/(short)0, acc, /*reuse_a=*/false, /*reuse_b=*/false);
        }
    }

    // add bias, then per-row sum of squares (row r lives in VGPR r%8, half r/8)
    const float bv = bias[n0 + nl];
    #pragma unroll
    for (int i = 0; i < 8; ++i) {
        float v = acc[i] + bv;
        acc[i] = v;
        float s = v * v;
        // reduce across the 16 lanes of this half (xor masks stay in-half)
        #pragma unroll
        for (int off = 1; off < 16; off <<= 1)
            s += __shfl_xor(s, off, 32);
        if (nl == 0) part[wave][i + 8 * half] = s;
    }
    __syncthreads();
    if (tid < 16) {
        float t = part[0][tid] + part[1][tid] + part[2][tid] + part[3][tid];
        rscale[tid] = 1.0f / fmaxf(sqrtf(t), 1e-12f);
    }
    __syncthreads();
    #pragma unroll
    for (int i = 0; i < 8; ++i) {
        int r = i + 8 * half;
        out[(size_t)(rowbase + r) * DIM + n0 + nl] = acc[i] * rscale[r];
    }
}

// out = ALPHA * in (elementwise, float4)
__global__ void scale_init(const float4* __restrict__ in, float4* __restrict__ out, int n4)
{
    int i = blockIdx.x * blockDim.x + threadIdx.x;
    int stride = gridDim.x * blockDim.x;
    for (; i < n4; i += stride) {
        float4 v = in[i];
        out[i] = make_float4(v.x * ALPHA, v.y * ALPHA, v.z * ALPHA, v.w * ALPHA);
    }
}

// scatter: out[dst[e]] += in[src[e]] * w[e]  (PyG 'add' aggregation)
// 16 threads per edge, each owns 4 consecutive dims (float4 gather).
__global__ void scatter_add(const float* __restrict__ in,
                            const int* __restrict__ src,
                            const int* __restrict__ dst,
                            const float* __restrict__ w,
                            float* __restrict__ out, int E)
{
    int t = blockIdx.x * blockDim.x + threadIdx.x;
    int e = t >> 4;
    const int q = (t & 15) << 2;
    const int estride = (gridDim.x * blockDim.x) >> 4;
    for (; e < E; e += estride) {
        const int s = src[e];
        const int d = dst[e];
        const float wt = w[e];
        const float4 v = *(const float4*)(in + (size_t)s * DIM + q);
        float* o = out + (size_t)d * DIM + q;
        atomicAdd(o + 0, v.x * wt);
        atomicAdd(o + 1, v.y * wt);
        atomicAdd(o + 2, v.z * wt);
        atomicAdd(o + 3, v.w * wt);
    }
}

extern "C" void kernel_launch(void* const* d_in, const int* in_sizes, int n_in,
                              void* d_out, int out_size, void* d_ws, size_t ws_size,
                              hipStream_t stream)
{
    const int*   ei    = (const int*)d_in[0];     // [2, E]
    const float* ew    = (const float*)d_in[1];   // [E, 1]
    const float* img_f = (const float*)d_in[2];   // [N_ITEMS, 2048]
    const float* txt_f = (const float*)d_in[3];   // [N_ITEMS, 384]
    const float* img_w = (const float*)d_in[4];   // [2048, 64]
    const float* img_b = (const float*)d_in[5];   // [64]
    const float* txt_w = (const float*)d_in[6];   // [384, 64]
    const float* txt_b = (const float*)d_in[7];   // [64]
    const float* img_p = (const float*)d_in[8];   // [N_USERS, 64]
    const float* txt_p = (const float*)d_in[9];   // [N_USERS, 64]
    float* out = (float*)d_out;                   // [2, N_NODES, 64]

    const int E = in_sizes[0] / 2;
    const int* src = ei;
    const int* dst = ei + E;

    float* bufA = (float*)d_ws;                        // layer input
    float* bufB = bufA + (size_t)N_NODES * DIM;        // layer-1 output

    const float* feats[2] = { img_f, txt_f };
    const float* wmat[2]  = { img_w, txt_w };
    const float* bvec[2]  = { img_b, txt_b };
    const float* pref[2]  = { img_p, txt_p };
    const int    Kdim[2]  = { 2048, 384 };

    const int n4 = N_NODES * DIM / 4;
    const int initBlocks = (n4 + 255) / 256;

    for (int m = 0; m < 2; ++m) {
        // item embeddings: l2norm(feats @ W + b) into rows [N_USERS, N_NODES)
        gemm_bias_l2norm_wmma<<<N_ITEMS / 16, 128, 0, stream>>>(
            feats[m], wmat[m], bvec[m], bufA + (size_t)N_USERS * DIM, Kdim[m]);
        // user preference rows [0, N_USERS)
        hipMemcpyAsync(bufA, pref[m], (size_t)N_USERS * DIM * sizeof(float),
                       hipMemcpyDeviceToDevice, stream);

        // layer 1: bufB = alpha*bufA + scatter(bufA)
        scale_init<<<initBlocks, 256, 0, stream>>>((const float4*)bufA, (float4*)bufB, n4);
        scatter_add<<<4096, 256, 0, stream>>>(bufA, src, dst, ew, bufB, E);

        // layer 2: d_out slab = alpha*bufB + scatter(bufB)
        float* dslab = out + (size_t)m * N_NODES * DIM;
        scale_init<<<initBlocks, 256, 0, stream>>>((const float4*)bufB, (float4*)dslab, n4);
        scatter_add<<<4096, 256, 0, stream>>>(bufB, src, dst, ew, dslab, E);
    }
}